// GCN_5102421148167
// MI455X (gfx1250) — compile-verified
//
#include <hip/hip_runtime.h>

// ---- problem constants (match reference) ----
#define NNODES 50000
#define NEDGES 800000
#define FDIM   64
#define HDIM   64
#define NBLK   3
#define NGRAPH 64
#define NCLS   10
#define BN_EPS 1e-5f

typedef __attribute__((ext_vector_type(2))) float v2f;
typedef __attribute__((ext_vector_type(8))) float v8f;

// ----------------------------------------------------------------------------
// small utility kernels
// ----------------------------------------------------------------------------
__global__ void k_zero(float* __restrict__ p, int n) {
    int i = blockIdx.x * blockDim.x + threadIdx.x;
    if (i < n) p[i] = 0.0f;
}

// count in-degree via hardware float atomics (exact for small counts)
__global__ void k_degree(const int* __restrict__ dst, float* __restrict__ deg, int E) {
    int e = blockIdx.x * blockDim.x + threadIdx.x;
    if (e < E) atomicAdd(&deg[dst[e]], 1.0f);
}

__global__ void k_dinv(float* __restrict__ deg, int n) {
    int i = blockIdx.x * blockDim.x + threadIdx.x;
    if (i < n) deg[i] = rsqrtf(deg[i] + 1.0f);
}

// ----------------------------------------------------------------------------
// BatchNorm statistics over prelu(h) : accumulate per-channel sum / sumsq
// blockDim = 256 (4 rows x 64 channels per block-iteration)
// ----------------------------------------------------------------------------
__global__ void k_bn_stats(const float* __restrict__ h,
                           const float* __restrict__ alpha_p,
                           float* __restrict__ acc /* [2*HDIM] */) {
    float a = *alpha_p;
    int c = threadIdx.x & (HDIM - 1);
    int r = blockIdx.x * (blockDim.x >> 6) + (threadIdx.x >> 6);
    int rstride = gridDim.x * (blockDim.x >> 6);
    float s = 0.0f, s2 = 0.0f;
    for (; r < NNODES; r += rstride) {
        float v = h[(size_t)r * HDIM + c];
        v = (v >= 0.0f) ? v : a * v;
        s += v; s2 += v * v;
    }
    atomicAdd(&acc[c], s);
    atomicAdd(&acc[HDIM + c], s2);
}

__global__ void k_bn_finalize(const float* __restrict__ acc,
                              const float* __restrict__ gamma,
                              const float* __restrict__ beta,
                              float* __restrict__ scale,
                              float* __restrict__ shift) {
    int c = threadIdx.x; // 64 threads
    float mean = acc[c] * (1.0f / NNODES);
    float var  = acc[HDIM + c] * (1.0f / NNODES) - mean * mean;
    float s = gamma[c] * rsqrtf(var + BN_EPS);
    scale[c] = s;
    shift[c] = beta[c] - mean * s;
}

// ----------------------------------------------------------------------------
// fp32 WMMA GEMM:  out[M,Nc] = T(A[M,K]) @ W[K,Nc] (+ obias)
// T = optional prelu(alpha) then optional per-channel affine (BatchNorm),
// selected at COMPILE TIME so the K-loop is straight-line code.
//
// One wave owns one 16-column stripe (ntile) and grid-strides over 16-row
// m-tiles. B fragments (and BN scale/shift fragments) for the whole K extent
// are preloaded into registers once — W is tiny and reused by all 3125
// m-tiles, so steady state is: one b64 A load + transform + v_wmma per K-step,
// fully unrolled (16 or 32 chained wmmas per tile).
//
// Fragment layouts per ISA 7.12.2 (V_WMMA_F32_16X16X4_F32):
//   A 16x4: lane<16 -> M=lane, {K+0,K+1}; lane>=16 -> {K+2,K+3}
//   B  4x16 symmetric; C/D: VGPR i holds M = i + 8*(lane>=16), N = lane&15.
// ----------------------------------------------------------------------------
template <int K, bool HASP, bool HASBN, bool HASBIAS>
__global__ __launch_bounds__(256)
void k_gemm_wmma(const float* __restrict__ A, const float* __restrict__ W,
                 float* __restrict__ out, int M, int Nc,
                 const float* __restrict__ alpha_p,
                 const float* __restrict__ scale,
                 const float* __restrict__ shift,
                 const float* __restrict__ obias) {
    constexpr int KF = K / 4;               // wmma steps along K
    int lane = threadIdx.x & 31;
    int wave = threadIdx.x >> 5;
    int lm   = lane & 15;
    int hs   = lane >> 4;                   // half-select (K+2/3, M+8)
    int ncol = blockIdx.y * 16 + lm;

    // ---- preload B fragments for the full K extent (registers) ----
    v2f bfrag[KF];
#pragma unroll
    for (int kf = 0; kf < KF; ++kf) {
        int c0 = kf * 4 + hs * 2;
        bfrag[kf] = v2f{W[(size_t)c0 * Nc + ncol], W[(size_t)(c0 + 1) * Nc + ncol]};
    }
    // ---- preload BN affine fragments (per input channel pair) ----
    v2f scf[HASBN ? KF : 1], shf[HASBN ? KF : 1];
    if (HASBN) {
#pragma unroll
        for (int kf = 0; kf < KF; ++kf) {
            int c0 = kf * 4 + hs * 2;
            scf[kf] = v2f{scale[c0], scale[c0 + 1]};
            shf[kf] = v2f{shift[c0], shift[c0 + 1]};
        }
    }
    float al = HASP ? *alpha_p : 0.0f;
    float ob = HASBIAS ? obias[ncol] : 0.0f;

    int MT = M >> 4;
    for (int mtile = blockIdx.x * 8 + wave; mtile < MT; mtile += gridDim.x * 8) {
        const float* arow = A + (size_t)(mtile * 16 + lm) * K + hs * 2;
        v8f acc = {0.f, 0.f, 0.f, 0.f, 0.f, 0.f, 0.f, 0.f};
#pragma unroll
        for (int kf = 0; kf < KF; ++kf) {
            v2f av = *(const v2f*)(arow + kf * 4);   // aligned b64 load
            if (HASP) {
                av.x = (av.x >= 0.0f) ? av.x : al * av.x;
                av.y = (av.y >= 0.0f) ? av.y : al * av.y;
            }
            if (HASBN) av = av * scf[kf] + shf[kf];
            acc = __builtin_amdgcn_wmma_f32_16x16x4_f32(
                      false, av, false, bfrag[kf], (short)0, acc, false, false);
        }
        float* orow = out + (size_t)(mtile * 16 + hs * 8) * Nc + ncol;
#pragma unroll
        for (int i = 0; i < 8; ++i)
            orow[(size_t)i * Nc] = acc[i] + ob;
    }
}

// ----------------------------------------------------------------------------
// aggregation: out = g * dinv^2 + bias  (self-loop term; also zero-inits out)
// ----------------------------------------------------------------------------
__global__ void k_agg_init(const float* __restrict__ g, const float* __restrict__ dinv,
                           const float* __restrict__ bias, float* __restrict__ out,
                           int cshift /* log2(C) */, int total) {
    int i = blockIdx.x * blockDim.x + threadIdx.x;
    if (i >= total) return;
    int n = i >> cshift;
    int c = i & ((1 << cshift) - 1);
    float d = dinv[n];
    out[i] = g[i] * (d * d) + bias[c];
}

// edge scatter: out[dst] += g[src] * dinv[src]*dinv[dst]  (float4 per thread)
__global__ void k_edge_scatter(const float* __restrict__ g, const int* __restrict__ ei,
                               const float* __restrict__ dinv, float* __restrict__ out,
                               int C, int gshift /* log2(C/4) */) {
    int t = blockIdx.x * blockDim.x + threadIdx.x;
    int total = NEDGES << gshift;
    if (t >= total) return;
    int e   = t >> gshift;
    int grp = t & ((1 << gshift) - 1);
    int src = ei[e];
    int dst = ei[NEDGES + e];
    float nrm = dinv[src] * dinv[dst];
    const float4 v = *(const float4*)(g + (size_t)src * C + grp * 4);
    float* o = out + (size_t)dst * C + grp * 4;
    atomicAdd(o + 0, v.x * nrm);
    atomicAdd(o + 1, v.y * nrm);
    atomicAdd(o + 2, v.z * nrm);
    atomicAdd(o + 3, v.w * nrm);
}

// ----------------------------------------------------------------------------
// per-graph mean pool (atomics), then tiny post-BN (one thread per channel)
// ----------------------------------------------------------------------------
__global__ void k_pool(const float* __restrict__ h, const int* __restrict__ batch,
                       float* __restrict__ pooled, float* __restrict__ cnt) {
    int i = blockIdx.x * blockDim.x + threadIdx.x;
    if (i >= NNODES * HDIM) return;
    int n = i >> 6;
    int c = i & 63;
    int b = batch[n];
    atomicAdd(&pooled[b * HDIM + c], h[i]);
    if (c == 0) atomicAdd(&cnt[b], 1.0f);
}

__global__ void k_postbn(const float* __restrict__ pooled, const float* __restrict__ cnt,
                         const float* __restrict__ gamma, const float* __restrict__ beta,
                         float* __restrict__ out) {
    int c = threadIdx.x; // 64 threads, one channel each
    float s = 0.0f, s2 = 0.0f;
    for (int g = 0; g < NGRAPH; ++g) {
        float v = pooled[g * HDIM + c] / fmaxf(cnt[g], 1.0f);
        s += v; s2 += v * v;
    }
    float mean = s * (1.0f / NGRAPH);
    float var  = s2 * (1.0f / NGRAPH) - mean * mean;
    float sc = gamma[c] * rsqrtf(var + BN_EPS);
    float sh = beta[c] - mean * sc;
    for (int g = 0; g < NGRAPH; ++g) {
        float v = pooled[g * HDIM + c] / fmaxf(cnt[g], 1.0f);
        out[g * HDIM + c] = v * sc + sh;
    }
}

// fc2: out[g,cls] = fc2_b + sum_k prelu(fc1[g,k]+fc1_b[k]) * fc2_w[k,cls]
__global__ void k_fc2(const float* __restrict__ f1, const float* __restrict__ f1b,
                      const float* __restrict__ alpha_p, const float* __restrict__ w2,
                      const float* __restrict__ b2, float* __restrict__ out) {
    int t = blockIdx.x * blockDim.x + threadIdx.x;
    if (t >= NGRAPH * NCLS) return;
    int g = t / NCLS;
    int cls = t - g * NCLS;
    float a = *alpha_p;
    float acc = b2[cls];
    for (int k = 0; k < 4 * HDIM; ++k) {
        float v = f1[g * 4 * HDIM + k] + f1b[k];
        v = (v >= 0.0f) ? v : a * v;
        acc += v * w2[k * NCLS + cls];
    }
    out[t] = acc;
}

// ----------------------------------------------------------------------------
// host orchestration
// ----------------------------------------------------------------------------
extern "C" void kernel_launch(void* const* d_in, const int* in_sizes, int n_in,
                              void* d_out, int out_size, void* d_ws, size_t ws_size,
                              hipStream_t stream) {
    const float* x          = (const float*)d_in[0];
    const int*   ei         = (const int*)  d_in[1];   // [2,E]
    const int*   batch      = (const int*)  d_in[2];
    const float* pre_w      = (const float*)d_in[3];
    const float* pre_b      = (const float*)d_in[4];
    const float* blk_prelu1 = (const float*)d_in[5];
    const float* bn_gamma   = (const float*)d_in[6];
    const float* bn_beta    = (const float*)d_in[7];
    const float* blk_w1     = (const float*)d_in[8];
    const float* blk_b1     = (const float*)d_in[9];
    const float* blk_prelu2 = (const float*)d_in[10];
    const float* blk_w2     = (const float*)d_in[11];
    const float* blk_b2     = (const float*)d_in[12];
    const float* post_gamma = (const float*)d_in[13];
    const float* post_beta  = (const float*)d_in[14];
    const float* fc1_w      = (const float*)d_in[15];
    const float* fc1_b      = (const float*)d_in[16];
    const float* post_prelu = (const float*)d_in[17];
    const float* fc2_w      = (const float*)d_in[18];
    const float* fc2_b      = (const float*)d_in[19];
    float* out = (float*)d_out;

    // workspace carve-out (256B aligned)
    char* ws = (char*)d_ws;
    size_t off = 0;
    auto alloc = [&](size_t bytes) -> void* {
        void* p = (void*)(ws + off);
        off += (bytes + 255) & ~(size_t)255;
        return p;
    };
    float* bufA   = (float*)alloc((size_t)NNODES * 128 * 4); // ping
    float* bufC   = (float*)alloc((size_t)NNODES * 128 * 4); // pong
    float* dinv   = (float*)alloc((size_t)NNODES * 4);
    float* bn_acc = (float*)alloc(2 * HDIM * 4);
    float* bn_sc  = (float*)alloc(HDIM * 4);
    float* bn_sh  = (float*)alloc(HDIM * 4);
    float* pooled = (float*)alloc(NGRAPH * HDIM * 4);
    float* cnt    = (float*)alloc(NGRAPH * 4);
    float* pbn    = (float*)alloc(NGRAPH * HDIM * 4);
    float* f1out  = (float*)alloc(NGRAPH * 4 * HDIM * 4);

    const int TPB = 256;
    // degrees -> dinv (shared across all 6 convs)
    k_zero<<<(NNODES + TPB - 1) / TPB, TPB, 0, stream>>>(dinv, NNODES);
    k_degree<<<(NEDGES + TPB - 1) / TPB, TPB, 0, stream>>>(ei + NEDGES, dinv, NEDGES);
    k_dinv<<<(NNODES + TPB - 1) / TPB, TPB, 0, stream>>>(dinv, NNODES);

    const int MT = NNODES / 16;              // 3125 m-tiles (exact)
    const int MWAVE = 4;                     // m-tiles per wave (amortize B preload)
    const int GX = (MT + 8 * MWAVE - 1) / (8 * MWAVE);
    dim3 gemmBlk(256);

    // pre: bufA = x @ pre_w + pre_b   [N,64]
    {
        dim3 grid(GX, FDIM / 16);
        k_gemm_wmma<64, false, false, true><<<grid, gemmBlk, 0, stream>>>(
            x, pre_w, bufA, NNODES, HDIM, nullptr, nullptr, nullptr, pre_b);
    }

    for (int i = 0; i < NBLK; ++i) {
        // BN stats over prelu(bufA)
        k_zero<<<1, 2 * HDIM, 0, stream>>>(bn_acc, 2 * HDIM);
        k_bn_stats<<<128, 256, 0, stream>>>(bufA, blk_prelu1 + i, bn_acc);
        k_bn_finalize<<<1, HDIM, 0, stream>>>(bn_acc, bn_gamma + i * HDIM,
                                              bn_beta + i * HDIM, bn_sc, bn_sh);
        // conv1 linear: bufC = BN(prelu(bufA)) @ w1[i]   [N,128]
        {
            dim3 grid(GX, (2 * HDIM) / 16);
            k_gemm_wmma<64, true, true, false><<<grid, gemmBlk, 0, stream>>>(
                bufA, blk_w1 + (size_t)i * HDIM * 2 * HDIM, bufC, NNODES, 2 * HDIM,
                blk_prelu1 + i, bn_sc, bn_sh, nullptr);
        }
        // aggregate (C=128): bufA = bufC*dinv^2 + b1 ; then += edge scatter
        k_agg_init<<<(NNODES * 128 + TPB - 1) / TPB, TPB, 0, stream>>>(
            bufC, dinv, blk_b1 + i * 2 * HDIM, bufA, 7, NNODES * 128);
        k_edge_scatter<<<((NEDGES << 5) + TPB - 1) / TPB, TPB, 0, stream>>>(
            bufC, ei, dinv, bufA, 128, 5);
        // conv2 linear: bufC = prelu(bufA) @ w2[i]   [N,64]
        {
            dim3 grid(GX, HDIM / 16);
            k_gemm_wmma<128, true, false, false><<<grid, gemmBlk, 0, stream>>>(
                bufA, blk_w2 + (size_t)i * 2 * HDIM * HDIM, bufC, NNODES, HDIM,
                blk_prelu2 + i, nullptr, nullptr, nullptr);
        }
        // aggregate (C=64)
        k_agg_init<<<(NNODES * 64 + TPB - 1) / TPB, TPB, 0, stream>>>(
            bufC, dinv, blk_b2 + i * HDIM, bufA, 6, NNODES * 64);
        k_edge_scatter<<<((NEDGES << 4) + TPB - 1) / TPB, TPB, 0, stream>>>(
            bufC, ei, dinv, bufA, 64, 4);
    }

    // mean pool + post BN
    k_zero<<<(NGRAPH * HDIM + TPB - 1) / TPB, TPB, 0, stream>>>(pooled, NGRAPH * HDIM);
    k_zero<<<1, NGRAPH, 0, stream>>>(cnt, NGRAPH);
    k_pool<<<(NNODES * HDIM + TPB - 1) / TPB, TPB, 0, stream>>>(bufA, batch, pooled, cnt);
    k_postbn<<<1, HDIM, 0, stream>>>(pooled, cnt, post_gamma, post_beta, pbn);

    // fc1 via WMMA: f1out = pbn @ fc1_w   [64,256]
    {
        dim3 grid(1, (4 * HDIM) / 16);   // 4 m-tiles live in the single block
        k_gemm_wmma<64, false, false, false><<<grid, gemmBlk, 0, stream>>>(
            pbn, fc1_w, f1out, NGRAPH, 4 * HDIM, nullptr, nullptr, nullptr, nullptr);
    }
    // fc2 (+fc1 bias, prelu fused)
    k_fc2<<<(NGRAPH * NCLS + 127) / 128, 128, 0, stream>>>(f1out, fc1_b, post_prelu,
                                                           fc2_w, fc2_b, out);
}